// EResGAT_73899207295409
// MI455X (gfx1250) — compile-verified
//
#include <hip/hip_runtime.h>
#include <math.h>

#define NN 100000
#define NE 1600000
#define F_IN 128
#define HEADS 4
#define F_OUT 16
#define C_TOT (HEADS * F_OUT) /* 64 */

typedef __attribute__((ext_vector_type(2))) float v2f;
typedef __attribute__((ext_vector_type(8))) float v8f;

// ---- monotone float <-> uint mapping for atomic max/min on signed floats ----
__device__ __forceinline__ unsigned ordF(float f) {
    unsigned b = __float_as_uint(f);
    return (b & 0x80000000u) ? ~b : (b | 0x80000000u);
}
__device__ __forceinline__ float unordF(unsigned u) {
    return (u & 0x80000000u) ? __uint_as_float(u & 0x7FFFFFFFu)
                             : __uint_as_float(~u);
}

// ---------------------------------------------------------------------------
// K0: zero out + denom, init ordered max/min cells
// ---------------------------------------------------------------------------
__global__ void init_ws(float* __restrict__ out, float* __restrict__ denom,
                        unsigned* __restrict__ redux) {
    long long i = (long long)blockIdx.x * blockDim.x + threadIdx.x;
    if (i < (long long)NN * C_TOT) out[i] = 0.0f;
    if (i < (long long)NN * HEADS) denom[i] = 0.0f;
    if (i == 0) { redux[0] = 0u; redux[1] = 0xFFFFFFFFu; }
}

// ---------------------------------------------------------------------------
// K1: proj[N,64] = x[N,128] @ W[128,64] via V_WMMA_F32_16X16X4_F32.
// One wave per 16x16 output tile; 4 waves/block cover all 4 heads of a
// 16-row slab (shared A rows hit L1/L2). K loop: 128/4 = 32 WMMAs per tile.
// Fragment layouts per CDNA5 ISA 7.12.2 (16x4 f32 A, 4x16 B, 16x16 f32 D).
// ---------------------------------------------------------------------------
__global__ __launch_bounds__(128) void gemm_proj(const float* __restrict__ x,
                                                 const float* __restrict__ W,
                                                 float* __restrict__ proj) {
    const int wave = threadIdx.x >> 5;     // head / 16-col tile: 0..3
    const int lane = threadIdx.x & 31;
    const int m0   = blockIdx.x * 16;      // row slab (exact: 100000 = 6250*16)
    const int n0   = wave * 16;
    const int idx  = lane & 15;            // A row / B col within tile
    const int kh   = (lane >> 4) * 2;      // K sub-offset: 0 (lanes 0-15) / 2

    const float* arow = x + (size_t)(m0 + idx) * F_IN + kh;
    const float* bcol = W + (size_t)kh * C_TOT + n0 + idx;

    v8f c = {0.f, 0.f, 0.f, 0.f, 0.f, 0.f, 0.f, 0.f};
    for (int kb = 0; kb < F_IN; kb += 4) {
        v2f a = *(const v2f*)(arow + kb);          // A[m, kb+kh], A[m, kb+kh+1]
        v2f b;
        b.x = bcol[(size_t)(kb + 0) * C_TOT];      // B[kb+kh,   n]
        b.y = bcol[(size_t)(kb + 1) * C_TOT];      // B[kb+kh+1, n]
        c = __builtin_amdgcn_wmma_f32_16x16x4_f32(
                /*neg_a=*/false, a, /*neg_b=*/false, b,
                /*c_mod=*/(short)0, c, /*reuse_a=*/false, /*reuse_b=*/false);
    }
    // D layout: lane holds col n0+(lane&15); VGPR j -> row m0 + j + (lane>>4)*8
    float* dst = proj + (size_t)(m0 + (lane >> 4) * 8) * C_TOT + n0 + idx;
#pragma unroll
    for (int j = 0; j < 8; ++j) dst[(size_t)j * C_TOT] = c[j];
}

// ---------------------------------------------------------------------------
// K2: per-(node,head) attention scores: s = <proj[n,h,:], a[h,:]>
// ---------------------------------------------------------------------------
__global__ void node_scores(const float* __restrict__ proj,
                            const float* __restrict__ a_src,
                            const float* __restrict__ a_trg,
                            float* __restrict__ s_src,
                            float* __restrict__ s_trg) {
    int i = blockIdx.x * blockDim.x + threadIdx.x;   // n*HEADS + h
    if (i >= NN * HEADS) return;
    int h = i & (HEADS - 1);
    const float* p  = proj + (size_t)i * F_OUT;
    const float* as = a_src + h * F_OUT;
    const float* at = a_trg + h * F_OUT;
    float ss = 0.f, st = 0.f;
#pragma unroll
    for (int f = 0; f < F_OUT; ++f) { float v = p[f]; ss += v * as[f]; st += v * at[f]; }
    s_src[i] = ss;
    s_trg[i] = st;
}

// ---------------------------------------------------------------------------
// K3: global max/min over all E*H leaky-relu scores (wave+LDS reduce,
// one ordered-uint atomic pair per 256-thread block).
// ---------------------------------------------------------------------------
__global__ __launch_bounds__(256) void edge_minmax(
        const long long* __restrict__ ei, const float* __restrict__ s_src,
        const float* __restrict__ s_trg, unsigned* __restrict__ redux) {
    int e = blockIdx.x * blockDim.x + threadIdx.x;
    float mx = -INFINITY, mn = INFINITY;
    if (e < NE) {
        int s = (int)ei[e];
        int t = (int)ei[(size_t)NE + e];
        float4 ssv = *(const float4*)(s_src + (size_t)s * HEADS);
        float4 stv = *(const float4*)(s_trg + (size_t)t * HEADS);
        float sc[4] = {ssv.x + stv.x, ssv.y + stv.y, ssv.z + stv.z, ssv.w + stv.w};
#pragma unroll
        for (int h = 0; h < 4; ++h) {
            float v = sc[h] > 0.f ? sc[h] : 0.2f * sc[h];
            mx = fmaxf(mx, v);
            mn = fminf(mn, v);
        }
    }
    for (int off = 16; off > 0; off >>= 1) {
        mx = fmaxf(mx, __shfl_down(mx, off, 32));
        mn = fminf(mn, __shfl_down(mn, off, 32));
    }
    __shared__ float smx[8], smn[8];
    int w = threadIdx.x >> 5, l = threadIdx.x & 31;
    if (l == 0) { smx[w] = mx; smn[w] = mn; }
    __syncthreads();
    if (threadIdx.x == 0) {
        float bmx = smx[0], bmn = smn[0];
        for (int k = 1; k < 8; ++k) { bmx = fmaxf(bmx, smx[k]); bmn = fminf(bmn, smn[k]); }
        atomicMax(&redux[0], ordF(bmx));
        atomicMin(&redux[1], ordF(bmn));
    }
}

// ---------------------------------------------------------------------------
// K4: softmax denominator (segment-sum of exp scores into denom[trg,h])
// ---------------------------------------------------------------------------
__global__ __launch_bounds__(256) void edge_denom(
        const long long* __restrict__ ei, const float* __restrict__ s_src,
        const float* __restrict__ s_trg, const unsigned* __restrict__ redux,
        float* __restrict__ denom) {
    int e = blockIdx.x * blockDim.x + threadIdx.x;
    if (e >= NE) return;
    float smax = unordF(redux[0]);
    float inv  = 3.0f / (smax - unordF(redux[1]));
    int s = (int)ei[e];
    int t = (int)ei[(size_t)NE + e];
    float4 ssv = *(const float4*)(s_src + (size_t)s * HEADS);
    float4 stv = *(const float4*)(s_trg + (size_t)t * HEADS);
    float sc[4] = {ssv.x + stv.x, ssv.y + stv.y, ssv.z + stv.z, ssv.w + stv.w};
#pragma unroll
    for (int h = 0; h < 4; ++h) {
        float v  = sc[h] > 0.f ? sc[h] : 0.2f * sc[h];
        float ex = __expf(inv * (v - smax));
        atomicAdd(&denom[(size_t)t * HEADS + h], ex);
    }
}

// ---------------------------------------------------------------------------
// K5: scatter weighted source features: 64 consecutive lanes = 64 channels
// of one edge -> coalesced proj reads and coalesced f32 atomics into out.
// All of proj/out/denom/scores live in the 192MB L2, so this runs at L2 rate.
// ---------------------------------------------------------------------------
__global__ __launch_bounds__(256) void edge_scatter(
        const long long* __restrict__ ei, const float* __restrict__ s_src,
        const float* __restrict__ s_trg, const float* __restrict__ proj,
        const float* __restrict__ denom, const unsigned* __restrict__ redux,
        float* __restrict__ out) {
    long long gid = (long long)blockIdx.x * blockDim.x + threadIdx.x;
    if (gid >= (long long)NE * C_TOT) return;
    int c = (int)(gid & (C_TOT - 1));
    int e = (int)(gid >> 6);
    int h = c >> 4;
    int s = (int)ei[e];
    int t = (int)ei[(size_t)NE + e];
    float sc = s_src[(size_t)s * HEADS + h] + s_trg[(size_t)t * HEADS + h];
    sc = sc > 0.f ? sc : 0.2f * sc;
    float smax = unordF(redux[0]);
    float ex   = __expf(3.0f * (sc - smax) / (smax - unordF(redux[1])));
    float att  = ex / (denom[(size_t)t * HEADS + h] + 1e-16f);
    atomicAdd(&out[(size_t)t * C_TOT + c], proj[(size_t)s * C_TOT + c] * att);
}

// ---------------------------------------------------------------------------
// K6: out = elu(out + bias)
// ---------------------------------------------------------------------------
__global__ void finalize(float* __restrict__ out, const float* __restrict__ bias) {
    int i = blockIdx.x * blockDim.x + threadIdx.x;
    if (i >= NN * C_TOT) return;
    float v = out[i] + bias[i & (C_TOT - 1)];
    out[i] = v > 0.f ? v : (__expf(v) - 1.0f);
}

extern "C" void kernel_launch(void* const* d_in, const int* in_sizes, int n_in,
                              void* d_out, int out_size, void* d_ws, size_t ws_size,
                              hipStream_t stream) {
    (void)in_sizes; (void)n_in; (void)out_size; (void)ws_size;
    const float*     x     = (const float*)d_in[0];
    const float*     W     = (const float*)d_in[1];
    const float*     a_src = (const float*)d_in[2];
    const float*     a_trg = (const float*)d_in[3];
    const float*     bias  = (const float*)d_in[4];
    const long long* ei    = (const long long*)d_in[5];   // int64 [2, E]
    float*           out   = (float*)d_out;

    // workspace layout (all 16B aligned): proj | s_src | s_trg | denom | redux
    float*    proj  = (float*)d_ws;
    float*    s_src = proj  + (size_t)NN * C_TOT;
    float*    s_trg = s_src + (size_t)NN * HEADS;
    float*    denom = s_trg + (size_t)NN * HEADS;
    unsigned* redux = (unsigned*)(denom + (size_t)NN * HEADS);

    init_ws   <<<(NN * C_TOT + 255) / 256, 256, 0, stream>>>(out, denom, redux);
    gemm_proj <<<NN / 16, 128, 0, stream>>>(x, W, proj);
    node_scores<<<(NN * HEADS + 255) / 256, 256, 0, stream>>>(proj, a_src, a_trg,
                                                              s_src, s_trg);
    edge_minmax<<<(NE + 255) / 256, 256, 0, stream>>>(ei, s_src, s_trg, redux);
    edge_denom <<<(NE + 255) / 256, 256, 0, stream>>>(ei, s_src, s_trg, redux, denom);
    edge_scatter<<<(int)(((long long)NE * C_TOT) / 256), 256, 0, stream>>>(
        ei, s_src, s_trg, proj, denom, redux, out);
    finalize  <<<(NN * C_TOT + 255) / 256, 256, 0, stream>>>(out, bias);
}